// EnhancedFinancialQuantumModel_50122268344522
// MI455X (gfx1250) — compile-verified
//
#include <hip/hip_runtime.h>
#include <math.h>

// Problem constants (from reference): B=8, S=2048, D=256, H=512, L=3, O=1
#define B_  8
#define S_  2048
#define D_  256
#define H_  512
#define L_  3
#define FF_ 2048
#define EPS_ 1e-8f

typedef __attribute__((ext_vector_type(2))) float v2f;
typedef __attribute__((ext_vector_type(4))) float v4f;   // native vector for NT stores
typedef __attribute__((ext_vector_type(8))) float v8f;

__device__ __forceinline__ float wave_reduce_sum(float v) {
#pragma unroll
  for (int off = 16; off > 0; off >>= 1) v += __shfl_xor(v, off, 32);
  return v;
}

// ---------------------------------------------------------------------------
// Kernel 1: fold proj_w / phase_emb into a 256x2 matrix + 2-vector bias.
// pc = (x @ proj_w.T + proj_b) @ phase_emb  ==  x @ W2 + bias2
// W2[d][t] = sum_k proj_w[k][d] * phase_emb[k][t]
// ---------------------------------------------------------------------------
__global__ void fold_phase_kernel(const float* __restrict__ proj_w,
                                  const float* __restrict__ proj_b,
                                  const float* __restrict__ phase_emb,
                                  float2* __restrict__ w2out /* [D_] + bias at [D_] */) {
  const int d = threadIdx.x;
  float a0 = 0.f, a1 = 0.f;
  for (int k = 0; k < D_; ++k) {
    const float pw = proj_w[k * D_ + d];
    a0 += pw * phase_emb[2 * k + 0];
    a1 += pw * phase_emb[2 * k + 1];
  }
  w2out[d] = make_float2(a0, a1);
  if (d == 0) {
    float b0 = 0.f, b1 = 0.f;
    for (int k = 0; k < D_; ++k) {
      b0 += proj_b[k] * phase_emb[2 * k + 0];
      b1 += proj_b[k] * phase_emb[2 * k + 1];
    }
    w2out[D_] = make_float2(b0, b1);
  }
}

// ---------------------------------------------------------------------------
// Kernel 2: psi. One wave (32 lanes) per (b,s) row: two length-256 dots,
// normalize, then 7 degenerate diffusion steps (mask==I  =>  force==0, each
// step is p *= r/(r+EPS)). Bandwidth-bound on reading x (16.8 MB).
// ---------------------------------------------------------------------------
__global__ void psi_kernel(const float* __restrict__ x,
                           const float2* __restrict__ w2,
                           float2* __restrict__ psi_out) {
  const int wave = threadIdx.x >> 5;
  const int lid  = threadIdx.x & 31;
  const int row  = blockIdx.x * 8 + wave;       // row in [0, B*S)
  const float* xr = x + (size_t)row * D_;
  float a0 = 0.f, a1 = 0.f;
#pragma unroll
  for (int j = 0; j < D_ / 32; ++j) {
    const int idx = lid + 32 * j;               // coalesced across the wave
    const float  xv = xr[idx];
    const float2 w  = w2[idx];
    a0 += xv * w.x;
    a1 += xv * w.y;
  }
  a0 = wave_reduce_sum(a0);
  a1 = wave_reduce_sum(a1);
  if (lid == 0) {
    const float2 bias = w2[D_];
    float p0 = a0 + bias.x, p1 = a1 + bias.y;
    const float n = sqrtf(p0 * p0 + p1 * p1);
    const float inv = 1.f / (n + EPS_);
    p0 *= inv; p1 *= inv;
    float r = sqrtf(p0 * p0 + p1 * p1);
#pragma unroll
    for (int it = 0; it < 7; ++it) {            // 3 steps dt=0.1 + 4 steps dt=0.05
      const float s = r / (r + EPS_);           // dt irrelevant: force == 0
      p0 *= s; p1 *= s; r *= s;
    }
    psi_out[row] = make_float2(p0, p1);
  }
}

// ---------------------------------------------------------------------------
// Kernel 3: mask = broadcast identity (134 MB). Pure streaming store —
// non-temporal so we don't trash the 192 MB L2 for the A kernel.
// ---------------------------------------------------------------------------
__global__ void mask_kernel(float* __restrict__ mask) {
  const int row = blockIdx.x;                   // b*S + s
  const int s   = row & (S_ - 1);
  float* mr = mask + (size_t)row * S_;
  const int c0 = threadIdx.x * 8;
  v4f v0, v1;
#pragma unroll
  for (int i = 0; i < 4; ++i) v0[i] = (c0 + i     == s) ? 1.f : 0.f;
#pragma unroll
  for (int i = 0; i < 4; ++i) v1[i] = (c0 + 4 + i == s) ? 1.f : 0.f;
  v4f* mp = (v4f*)(mr + c0);
  __builtin_nontemporal_store(v0, mp);
  __builtin_nontemporal_store(v1, mp + 1);
}

// ---------------------------------------------------------------------------
// Kernel 4: A[b] = psi[b] @ psi[b].T  — rank-2 outer product, 134 MB of
// streaming stores. psi[b] (16 KB) staged once in LDS per block.
// ---------------------------------------------------------------------------
__global__ void gram_kernel(const float2* __restrict__ psi, float* __restrict__ A) {
  __shared__ float2 ps[S_];                     // 16 KB
  const int b = blockIdx.y;
  const float2* pb = psi + (size_t)b * S_;
  for (int i = threadIdx.x; i < S_; i += 256) ps[i] = pb[i];
  __syncthreads();
  const int s0 = blockIdx.x * 8;
  float* Ab = A + ((size_t)b * S_ + s0) * S_;
  const int c0 = threadIdx.x * 8;
#pragma unroll
  for (int r = 0; r < 8; ++r) {
    const float2 pr = ps[s0 + r];
    v4f v0, v1;
#pragma unroll
    for (int i = 0; i < 4; ++i) {
      const float2 pc = ps[c0 + i];
      v0[i] = pr.x * pc.x + pr.y * pc.y;
    }
#pragma unroll
    for (int i = 0; i < 4; ++i) {
      const float2 pc = ps[c0 + 4 + i];
      v1[i] = pr.x * pc.x + pr.y * pc.y;
    }
    v4f* op = (v4f*)(Ab + (size_t)r * S_ + c0);
    __builtin_nontemporal_store(v0, op);
    __builtin_nontemporal_store(v1, op + 1);
  }
}

// ---------------------------------------------------------------------------
// WMMA f32 16x16x4 GEMM helper. C(16xN) = A_lds(16xK) @ W.T (+bias, +silu),
// W is (N,K) row-major in global memory. Fragment layouts per ISA 7.12.2:
//   A 16x4 : lane m (0-15) VGPR{0,1} = A[m][k0+{0,1}];  lanes 16-31 = K+2
//   B 4x16 : lane m VGPR{0,1} = B[k0+{0,1}][n] = W[n][k0+{0,1}]; halves K+2
//   C 16x16: VGPR g: lanes 0-15 -> M=g, lanes 16-31 -> M=g+8, N=lane&15
// ---------------------------------------------------------------------------
__device__ __forceinline__ void gemm16_wmma(const float* __restrict__ Alds, int K,
                                            const float* __restrict__ W,
                                            const float* __restrict__ bias,
                                            float* __restrict__ Clds, int N,
                                            int silu_act) {
  const int wave   = threadIdx.x >> 5;
  const int nwaves = blockDim.x >> 5;
  const int lid    = threadIdx.x & 31;
  const int half   = lid >> 4;
  const int m      = lid & 15;
  for (int nt = wave * 16; nt < N; nt += nwaves * 16) {
    v8f acc = {0.f, 0.f, 0.f, 0.f, 0.f, 0.f, 0.f, 0.f};
    const int n = nt + m;
    const float* wr = W + (size_t)n * K;
    for (int k0 = 0; k0 < K; k0 += 4) {
      const int ks = k0 + 2 * half;
      v2f a, bf;
      a.x  = Alds[m * K + ks];
      a.y  = Alds[m * K + ks + 1];
      bf.x = wr[ks];
      bf.y = wr[ks + 1];
      acc = __builtin_amdgcn_wmma_f32_16x16x4_f32(false, a, false, bf,
                                                  (short)0, acc, false, false);
    }
    const float bs = bias ? bias[n] : 0.f;
#pragma unroll
    for (int g = 0; g < 8; ++g) {
      const int row = g + 8 * half;
      float v = acc[g] + bs;
      if (silu_act) v = v / (1.f + __expf(-v));   // silu = x*sigmoid(x)
      Clds[(size_t)row * N + n] = v;
    }
  }
}

// LayerNorm over rows held in LDS: dst = LN(a (+ badd)) * w + bias.
__device__ __forceinline__ void ln_rows(float* __restrict__ dst,
                                        const float* __restrict__ a,
                                        const float* __restrict__ badd,
                                        const float* __restrict__ w,
                                        const float* __restrict__ bias,
                                        int ncols, int nrows) {
  const int wave = threadIdx.x >> 5, nwaves = blockDim.x >> 5, lid = threadIdx.x & 31;
  for (int r = wave; r < nrows; r += nwaves) {
    float sum = 0.f, sq = 0.f;
    for (int c = lid; c < ncols; c += 32) {
      const float t = a[r * ncols + c] + (badd ? badd[r * ncols + c] : 0.f);
      dst[r * ncols + c] = t;
      sum += t; sq += t * t;
    }
    sum = wave_reduce_sum(sum);
    sq  = wave_reduce_sum(sq);
    const float mean = sum / ncols;
    const float var  = sq / ncols - mean * mean;
    const float inv  = rsqrtf(var + 1e-5f);
    for (int c = lid; c < ncols; c += 32) {
      const float t = dst[r * ncols + c];
      dst[r * ncols + c] = (t - mean) * inv * w[c] + bias[c];
    }
  }
}

// ---------------------------------------------------------------------------
// Kernel 5: the whole transformer head in ONE workgroup (16 waves, 224 KB LDS
// out of the WGP's 320 KB). Batch 8 padded to M=16 for WMMA. Seq-len is 1 so
// softmax==1 and o==v: q/k projections are dead code and skipped entirely.
// ---------------------------------------------------------------------------
__global__ void head_kernel(const float2* __restrict__ psi,
                            const float* __restrict__ hebb,
                            const float* __restrict__ ainw, const float* __restrict__ ainb,
                            const float* __restrict__ aoutw, const float* __restrict__ aoutb,
                            const float* __restrict__ f1w,  const float* __restrict__ f1b,
                            const float* __restrict__ f2w,  const float* __restrict__ f2b,
                            const float* __restrict__ l1w,  const float* __restrict__ l1b,
                            const float* __restrict__ l2w,  const float* __restrict__ l2b,
                            const float* __restrict__ nAw,  const float* __restrict__ nAb,
                            const float* __restrict__ nBw,  const float* __restrict__ nBb,
                            const float* __restrict__ p1w,  const float* __restrict__ p1b,
                            const float* __restrict__ p2w,  const float* __restrict__ p2b,
                            float* __restrict__ preds) {
  extern __shared__ float smem[];
  float* hbuf = smem;               // 16 x 512  (32 KB)
  float* vbuf = hbuf + 16 * H_;     // 16 x 512  (32 KB)
  float* obuf = vbuf + 16 * H_;     // 16 x 512  (32 KB)
  float* fbuf = obuf + 16 * H_;     // 16 x 2048 (128 KB)
  const int tid = threadIdx.x;

  // h = final_psi @ hebb_w.T ; rows 8..15 zero-padded.
  for (int idx = tid; idx < 16 * H_; idx += blockDim.x) {
    const int row = idx >> 9, j = idx & (H_ - 1);
    float v = 0.f;
    if (row < B_) {
      const float2 fp = psi[(size_t)row * S_ + (S_ - 1)];
      v = fp.x * hebb[2 * j] + fp.y * hebb[2 * j + 1];
    }
    hbuf[idx] = v;
  }
  __syncthreads();

  for (int l = 0; l < L_; ++l) {
    const float* Wv = ainw + (size_t)l * 3 * H_ * H_ + (size_t)2 * H_ * H_;  // v slice
    const float* bv = ainb + (size_t)l * 3 * H_ + 2 * H_;
    const float* Wo = aoutw + (size_t)l * H_ * H_;
    const float* bo = aoutb + (size_t)l * H_;
    gemm16_wmma(hbuf, H_, Wv, bv, vbuf, H_, 0);                  __syncthreads();
    gemm16_wmma(vbuf, H_, Wo, bo, obuf, H_, 0);                  __syncthreads();
    ln_rows(hbuf, hbuf, obuf, l1w + l * H_, l1b + l * H_, H_, 16); __syncthreads();
    gemm16_wmma(hbuf, H_, f1w + (size_t)l * FF_ * H_, f1b + l * FF_, fbuf, FF_, 1);
    __syncthreads();
    gemm16_wmma(fbuf, FF_, f2w + (size_t)l * H_ * FF_, f2b + l * H_, obuf, H_, 0);
    __syncthreads();
    ln_rows(hbuf, hbuf, obuf, l2w + l * H_, l2b + l * H_, H_, 16); __syncthreads();
  }

  ln_rows(hbuf, hbuf, nullptr, nAw, nAb, H_, 16);                __syncthreads();

  // p = silu(h @ p1_w.T + b) : 8 x 256, one wave per output element.
  {
    const int wave = tid >> 5, lid = tid & 31, nwaves = blockDim.x >> 5;
    for (int i = wave; i < B_ * 256; i += nwaves) {
      const int r = i >> 8, j = i & 255;
      const float* wrow = p1w + (size_t)j * H_;
      float acc = 0.f;
      for (int c = lid; c < H_; c += 32) acc += hbuf[r * H_ + c] * wrow[c];
      acc = wave_reduce_sum(acc);
      if (lid == 0) {
        const float v = acc + p1b[j];
        vbuf[r * 256 + j] = v / (1.f + __expf(-v));
      }
    }
  }
  __syncthreads();
  ln_rows(vbuf, vbuf, nullptr, nBw, nBb, 256, B_);               __syncthreads();

  // preds = p @ p2_w.T + p2_b : one wave per batch row.
  {
    const int wave = tid >> 5, lid = tid & 31;
    if (wave < B_) {
      float acc = 0.f;
      for (int c = lid; c < 256; c += 32) acc += vbuf[wave * 256 + c] * p2w[c];
      acc = wave_reduce_sum(acc);
      if (lid == 0) preds[wave] = acc + p2b[0];
    }
  }
}

// ---------------------------------------------------------------------------
extern "C" void kernel_launch(void* const* d_in, const int* in_sizes, int n_in,
                              void* d_out, int out_size, void* d_ws, size_t ws_size,
                              hipStream_t stream) {
  const float* x        = (const float*)d_in[0];
  const float* proj_w   = (const float*)d_in[1];
  const float* proj_b   = (const float*)d_in[2];
  const float* phase    = (const float*)d_in[3];
  const float* hebb     = (const float*)d_in[4];
  const float* ainw     = (const float*)d_in[5];
  const float* ainb     = (const float*)d_in[6];
  const float* aoutw    = (const float*)d_in[7];
  const float* aoutb    = (const float*)d_in[8];
  const float* f1w      = (const float*)d_in[9];
  const float* f1b      = (const float*)d_in[10];
  const float* f2w      = (const float*)d_in[11];
  const float* f2b      = (const float*)d_in[12];
  const float* l1w      = (const float*)d_in[13];
  const float* l1b      = (const float*)d_in[14];
  const float* l2w      = (const float*)d_in[15];
  const float* l2b      = (const float*)d_in[16];
  const float* nAw      = (const float*)d_in[17];
  const float* nAb      = (const float*)d_in[18];
  const float* nBw      = (const float*)d_in[19];
  const float* nBb      = (const float*)d_in[20];
  const float* p1w      = (const float*)d_in[21];
  const float* p1b      = (const float*)d_in[22];
  const float* p2w      = (const float*)d_in[23];
  const float* p2b      = (const float*)d_in[24];

  // Output tuple layout: preds (8) | A (B*S*S) | mask (B*S*S) | psi (B*S*2)
  float* out   = (float*)d_out;
  float* preds = out;
  float* A     = out + B_;                               // B*O == 8
  float* mask  = A + (size_t)B_ * S_ * S_;
  float* psi   = mask + (size_t)B_ * S_ * S_;

  float2* w2 = (float2*)d_ws;                            // 257 float2 in scratch

  fold_phase_kernel<<<1, D_, 0, stream>>>(proj_w, proj_b, phase, w2);
  psi_kernel<<<(B_ * S_) / 8, 256, 0, stream>>>(x, w2, (float2*)psi);
  mask_kernel<<<B_ * S_, 256, 0, stream>>>(mask);
  gram_kernel<<<dim3(S_ / 8, B_), 256, 0, stream>>>((const float2*)psi, A);

  const size_t smem_bytes = (size_t)(16 * H_ * 3 + 16 * FF_) * sizeof(float); // 224 KB
  (void)hipFuncSetAttribute((const void*)head_kernel,
                            hipFuncAttributeMaxDynamicSharedMemorySize,
                            (int)smem_bytes);
  head_kernel<<<1, 512, smem_bytes, stream>>>((const float2*)psi, hebb,
                                              ainw, ainb, aoutw, aoutb,
                                              f1w, f1b, f2w, f2b,
                                              l1w, l1b, l2w, l2b,
                                              nAw, nAb, nBw, nBb,
                                              p1w, p1b, p2w, p2b, preds);
}